// NNConvBlock_3487513444356
// MI455X (gfx1250) — compile-verified
//
#include <hip/hip_runtime.h>

// NNConv + GraphNorm fused pipeline for MI455X (gfx1250, wave32).
//
// Refactor: msg[e] = h_e @ P_{src(e)} + c_{src(e)},
//   P_n[k,o] = sum_i x_n[i] * W2[k, i*32+o]   (node GEMM, 13.1 GFLOP via fp32 WMMA)
//   c_n[o]   = sum_i x_n[i] * b2[i*32+o]
// P for a 16-node tile (128KB) lives entirely in LDS; edges are grouped by
// source node via a CSR built on-device so the producing workgroup consumes
// its own P tile.

typedef float v2f __attribute__((ext_vector_type(2)));
typedef float v8f __attribute__((ext_vector_type(8)));

#define DIM 32
#define HID 64
#define EIN 16
#define NG  1000       // graphs (G)
#define NB  16         // nodes per block in main kernel

// ---------------- utility kernels ----------------

__global__ void k_zero(float* __restrict__ p, long n) {
  long stride = (long)gridDim.x * blockDim.x;
  for (long i = (long)blockIdx.x * blockDim.x + threadIdx.x; i < n; i += stride)
    p[i] = 0.0f;
}

// W2p[i*2048 + k*32 + o] = W2[k*1024 + i*32 + o]
__global__ void k_permW2(const float* __restrict__ W2, float* __restrict__ W2p) {
  int t = blockIdx.x * blockDim.x + threadIdx.x;
  if (t >= HID * DIM * DIM) return;
  int k = t >> 10;            // 0..63
  int col = t & 1023;         // i*32+o
  int i = col >> 5, o = col & 31;
  W2p[i * (HID * DIM) + k * DIM + o] = W2[t];
}

// out-degree histogram by src (for CSR) + in-degree by dst (for scatter-mean)
__global__ void k_hist(const int* __restrict__ eidx, unsigned* __restrict__ hist,
                       float* __restrict__ deg, int E) {
  int e = blockIdx.x * blockDim.x + threadIdx.x;
  if (e >= E) return;
  atomicAdd(&hist[eidx[e]], 1u);
  atomicAdd(&deg[eidx[E + e]], 1.0f);
}

// single-block exclusive scan of hist[0..N) -> rowptr[0..N]
__global__ void k_scan(const unsigned* __restrict__ hist,
                       unsigned* __restrict__ rowptr, int N) {
  __shared__ unsigned sd[1024];
  __shared__ unsigned soff;
  int tid = threadIdx.x;
  if (tid == 0) soff = 0;
  __syncthreads();
  for (int base = 0; base < N; base += 1024) {
    int i = base + tid;
    unsigned v = (i < N) ? hist[i] : 0u;
    sd[tid] = v;
    __syncthreads();
    for (int off = 1; off < 1024; off <<= 1) {
      unsigned t = (tid >= off) ? sd[tid - off] : 0u;
      __syncthreads();
      sd[tid] += t;
      __syncthreads();
    }
    if (i < N) rowptr[i] = soff + sd[tid] - v;   // exclusive
    __syncthreads();
    if (tid == 0) soff += sd[1023];
    __syncthreads();
  }
  if (tid == 0) rowptr[N] = soff;
}

__global__ void k_scatter(const int* __restrict__ eidx,
                          const unsigned* __restrict__ rowptr,
                          unsigned* __restrict__ cursor,
                          unsigned* __restrict__ order, int E) {
  int e = blockIdx.x * blockDim.x + threadIdx.x;
  if (e >= E) return;
  int s = eidx[e];
  unsigned pos = rowptr[s] + atomicAdd(&cursor[s], 1u);
  order[pos] = (unsigned)e;
}

// ---------------- main fused kernel: node WMMA + edge messages ----------------

__global__ __launch_bounds__(256) void k_main(
    const float* __restrict__ x, const float* __restrict__ edge_attr,
    const float* __restrict__ W1, const float* __restrict__ b1,
    const float* __restrict__ b2, const float* __restrict__ W2p,
    const unsigned* __restrict__ rowptr, const unsigned* __restrict__ order,
    const int* __restrict__ eidx, float* __restrict__ aggsum, int N, int E) {
  __shared__ float Pl[NB * HID * DIM];  // 128KB: P[m][k*32+o]
  __shared__ float Xt[NB * DIM];        // 2KB
  __shared__ float W1s[EIN * HID];      // 4KB
  __shared__ float b1s[HID];
  __shared__ float b2s[DIM * DIM];      // 4KB
  __shared__ float cB[NB * DIM];        // 2KB

  const int tid = threadIdx.x;
  const int lane = tid & 31;
  const int wave = tid >> 5;
  const int base = blockIdx.x * NB;

  for (int t = tid; t < NB * DIM; t += 256) {
    int n = base + (t >> 5);
    Xt[t] = (n < N) ? x[n * DIM + (t & 31)] : 0.0f;
  }
  for (int t = tid; t < EIN * HID; t += 256) W1s[t] = W1[t];
  if (tid < HID) b1s[tid] = b1[tid];
  for (int t = tid; t < DIM * DIM; t += 256) b2s[t] = b2[t];
  __syncthreads();

  // ---- WMMA phase: P16[16 x 2048] = Xt[16 x 32] @ W2p[32 x 2048] (fp32) ----
  // A 16x4 f32 layout: lanes 0-15 -> M=0..15; VGPR0 holds K=kb+2*half,
  // VGPR1 holds K+1 (half = lane>=16). C layout: VGPR j -> M=j (+8 for half).
  const int m = lane & 15;
  const int hsel = lane >> 4;
  v2f a[8];
#pragma unroll
  for (int s = 0; s < 8; ++s) {
    int kk = s * 4 + 2 * hsel;
    a[s].x = Xt[m * DIM + kk];
    a[s].y = Xt[m * DIM + kk + 1];
  }
  for (int nt = wave * 16; nt < wave * 16 + 16; ++nt) {  // 128 n-tiles / 8 waves
    const int ncol = nt * 16 + m;
    v8f acc = {0.f, 0.f, 0.f, 0.f, 0.f, 0.f, 0.f, 0.f};
#pragma unroll
    for (int s = 0; s < 8; ++s) {
      int kk = s * 4 + 2 * hsel;
      v2f b;
      b.x = W2p[kk * (HID * DIM) + ncol];
      b.y = W2p[(kk + 1) * (HID * DIM) + ncol];
      acc = __builtin_amdgcn_wmma_f32_16x16x4_f32(
          false, a[s], false, b, (short)0, acc, false, false);
    }
#pragma unroll
    for (int j = 0; j < 8; ++j)
      Pl[(j + 8 * hsel) * (HID * DIM) + ncol] = acc[j];
  }

  // c_n[o] = sum_i x_n[i] * b2[i*32+o]
  for (int t = tid; t < NB * DIM; t += 256) {
    int ln = t >> 5, o = t & 31;
    float c = 0.f;
#pragma unroll
    for (int i = 0; i < DIM; ++i) c = fmaf(Xt[ln * DIM + i], b2s[i * DIM + o], c);
    cB[t] = c;
  }
  __syncthreads();

  // ---- edge phase: this block's nodes own edges [rowptr[base], rowptr[hi]) ----
  int hi = min(base + NB, N);
  unsigned ebeg = rowptr[base], eend = rowptr[hi];
  for (unsigned j = ebeg + (unsigned)wave; j < eend; j += 8) {
    int e = (int)order[j];
    int s = eidx[e];
    int d = eidx[E + e];
    int local = s - base;
    // h = relu(attr @ W1 + b1); lane computes h[lane], h[lane+32]
    float h0 = b1s[lane], h1 = b1s[lane + 32];
#pragma unroll
    for (int i = 0; i < EIN; ++i) {
      float av = edge_attr[e * EIN + i];  // uniform across lanes -> broadcast
      h0 = fmaf(av, W1s[i * HID + lane], h0);
      h1 = fmaf(av, W1s[i * HID + lane + 32], h1);
    }
    h0 = fmaxf(h0, 0.f);
    h1 = fmaxf(h1, 0.f);
    // msg[o=lane] = c + sum_k h[k] * P[local][k][lane]
    float msg = cB[local * DIM + lane];
    const float* Pn = &Pl[local * (HID * DIM) + lane];
#pragma unroll
    for (int k = 0; k < 32; ++k) msg = fmaf(__shfl(h0, k, 32), Pn[k * DIM], msg);
#pragma unroll
    for (int k = 0; k < 32; ++k)
      msg = fmaf(__shfl(h1, k, 32), Pn[(k + 32) * DIM], msg);
    atomicAdd(&aggsum[d * DIM + lane], msg);
  }
}

// ---------------- node update + graph statistics ----------------

__global__ __launch_bounds__(256) void k_node(
    const float* __restrict__ x, const float* __restrict__ aggsum,
    const float* __restrict__ deg, const float* __restrict__ W_root,
    const float* __restrict__ conv_bias, const int* __restrict__ batch,
    float* __restrict__ out_ws, float* __restrict__ gsum,
    float* __restrict__ gsumsq, float* __restrict__ gcnt, int N) {
  int tid = blockIdx.x * 256 + threadIdx.x;
  int n = tid >> 5, lane = tid & 31;
  if (n >= N) return;
  float xi = x[n * DIM + lane];
  float acc = 0.f;
#pragma unroll
  for (int i = 0; i < DIM; ++i)
    acc = fmaf(__shfl(xi, i, 32), W_root[i * DIM + lane], acc);
  float dg = fmaxf(deg[n], 1.0f);
  float o = aggsum[n * DIM + lane] / dg + acc + conv_bias[lane];
  o = fmaxf(o, 0.f);
  out_ws[n * DIM + lane] = o;
  int g = batch[n];
  atomicAdd(&gsum[g * DIM + lane], o);
  atomicAdd(&gsumsq[g * DIM + lane], o * o);
  if (lane == 0) atomicAdd(&gcnt[g], 1.0f);
}

__global__ void k_gstats(const float* __restrict__ gsum,
                         const float* __restrict__ gsumsq,
                         const float* __restrict__ gcnt,
                         const float* __restrict__ gnw,
                         const float* __restrict__ gnb,
                         const float* __restrict__ gnms,
                         float* __restrict__ scaleA, float* __restrict__ biasB) {
  int tid = blockIdx.x * blockDim.x + threadIdx.x;
  if (tid >= NG * DIM) return;
  int g = tid >> 5, o = tid & 31;
  float cnt = fmaxf(gcnt[g], 1.0f);
  float mean = gsum[tid] / cnt;
  float msc = mean * gnms[o];
  // var = E[(out - msc)^2] = E[out^2] - 2*msc*mean + msc^2
  float var = gsumsq[tid] / cnt - 2.0f * msc * mean + msc * msc;
  float inv = rsqrtf(var + 1e-5f);
  float A = gnw[o] * inv;
  scaleA[tid] = A;
  biasB[tid] = gnb[o] - A * msc;
}

__global__ void k_final(const float* __restrict__ x,
                        const float* __restrict__ out_ws,
                        const float* __restrict__ scaleA,
                        const float* __restrict__ biasB,
                        const int* __restrict__ batch, float* __restrict__ out,
                        int N) {
  int tid = blockIdx.x * blockDim.x + threadIdx.x;
  if (tid >= N * DIM) return;
  int n = tid >> 5, o = tid & 31;
  int g = batch[n];
  out[tid] = fmaf(scaleA[g * DIM + o], out_ws[tid], biasB[g * DIM + o]) + x[tid];
}

// ---------------- launch ----------------

extern "C" void kernel_launch(void* const* d_in, const int* in_sizes, int n_in,
                              void* d_out, int out_size, void* d_ws,
                              size_t ws_size, hipStream_t stream) {
  const float* x         = (const float*)d_in[0];
  const float* edge_attr = (const float*)d_in[1];
  const float* W1        = (const float*)d_in[2];
  const float* b1        = (const float*)d_in[3];
  const float* W2        = (const float*)d_in[4];
  const float* b2        = (const float*)d_in[5];
  const float* W_root    = (const float*)d_in[6];
  const float* conv_bias = (const float*)d_in[7];
  const float* gnw       = (const float*)d_in[8];
  const float* gnb       = (const float*)d_in[9];
  const float* gnms      = (const float*)d_in[10];
  const int*   eidx      = (const int*)d_in[11];   // [2,E]
  const int*   batch     = (const int*)d_in[12];

  const int N = in_sizes[0] / DIM;
  const int E = in_sizes[11] / 2;

  // workspace carve-up (256B aligned)
  size_t off = 0;
  auto alloc = [&](size_t bytes) -> char* {
    off = (off + 255) & ~(size_t)255;
    char* p = (char*)d_ws + off;
    off += bytes;
    return p;
  };
  float*    aggsum = (float*)alloc((size_t)N * DIM * 4);
  float*    deg    = (float*)alloc((size_t)N * 4);
  float*    gsum   = (float*)alloc((size_t)NG * DIM * 4);
  float*    gsumsq = (float*)alloc((size_t)NG * DIM * 4);
  float*    gcnt   = (float*)alloc((size_t)NG * 4);
  unsigned* hist   = (unsigned*)alloc((size_t)N * 4);
  unsigned* cursor = (unsigned*)alloc((size_t)N * 4);
  size_t zero_end = (off + 255) & ~(size_t)255;   // everything above is zeroed
  unsigned* rowptr = (unsigned*)alloc((size_t)(N + 1) * 4);
  unsigned* order  = (unsigned*)alloc((size_t)E * 4);
  float*    W2p    = (float*)alloc((size_t)DIM * HID * DIM * 4);
  float*    out_ws = (float*)alloc((size_t)N * DIM * 4);
  float*    scaleA = (float*)alloc((size_t)NG * DIM * 4);
  float*    biasB  = (float*)alloc((size_t)NG * DIM * 4);

  long zero_words = (long)(zero_end / 4);
  k_zero<<<2048, 256, 0, stream>>>((float*)d_ws, zero_words);

  k_permW2<<<(HID * DIM * DIM + 255) / 256, 256, 0, stream>>>(W2, W2p);
  k_hist<<<(E + 255) / 256, 256, 0, stream>>>(eidx, hist, deg, E);
  k_scan<<<1, 1024, 0, stream>>>(hist, rowptr, N);
  k_scatter<<<(E + 255) / 256, 256, 0, stream>>>(eidx, rowptr, cursor, order, E);

  k_main<<<(N + NB - 1) / NB, 256, 0, stream>>>(
      x, edge_attr, W1, b1, b2, W2p, rowptr, order, eidx, aggsum, N, E);

  k_node<<<(N * DIM + 255) / 256, 256, 0, stream>>>(
      x, aggsum, deg, W_root, conv_bias, batch, out_ws, gsum, gsumsq, gcnt, N);

  k_gstats<<<(NG * DIM + 255) / 256, 256, 0, stream>>>(
      gsum, gsumsq, gcnt, gnw, gnb, gnms, scaleA, biasB);

  k_final<<<(N * DIM + 255) / 256, 256, 0, stream>>>(
      x, out_ws, scaleA, biasB, batch, (float*)d_out, N);
}